// Custom_Attention_88192858456832
// MI455X (gfx1250) — compile-verified
//
#include <hip/hip_runtime.h>

typedef __attribute__((ext_vector_type(2))) float v2f;
typedef __attribute__((ext_vector_type(8))) float v8f;

// Problem shape (from reference): x [B=64, L=2048] fp32 -> y [B, L] fp32.
// y[b,i] = sum_j x_j*exp(x_i*x_j/16) / sum_j exp(x_i*x_j/16)
// (softmax max-shift cancels in the ratio; |x_i*x_j/16| <~ 1.3, no overflow)

#define L_DIM 2048
#define WAVES_PER_BLOCK 8
#define ROWS_PER_BLOCK (WAVES_PER_BLOCK * 16)   // 128

#define LOG2E 1.4426950408889634f

__global__ __launch_bounds__(256) void rank1_softmax_attn_kernel(
    const float* __restrict__ x, float* __restrict__ y, int B)
{
    __shared__ float sx[L_DIM];                    // one batch row of x (8 KB)
    __shared__ float sred[WAVES_PER_BLOCK][32];    // per-wave num[16] + den[16]

    const int tid  = threadIdx.x;
    const int wave = tid >> 5;
    const int lane = tid & 31;

    const int blocksPerBatch = L_DIM / ROWS_PER_BLOCK;       // 16
    const int b   = blockIdx.x / blocksPerBatch;
    const int rb  = blockIdx.x % blocksPerBatch;
    const int row0 = rb * ROWS_PER_BLOCK + wave * 16;        // this wave's 16-row tile

    // Cooperative load of x[b, :] into LDS as float4 (2 per thread).
    {
        const float4* xg = (const float4*)(x + (size_t)b * L_DIM);
        float4* sx4 = (float4*)sx;
        sx4[tid]       = xg[tid];
        sx4[tid + 256] = xg[tid + 256];
    }
    __syncthreads();

    // WMMA A-matrix layout (16x4 f32): lanes 0-15 hold row M=lane, K={0,1};
    // lanes 16-31 hold row M=lane-16, K={2,3}.
    const int   mi    = lane & 15;        // row within tile / column of B
    const int   khalf = lane >> 4;        // 0 -> K pair {0,1}; 1 -> K pair {2,3}
    // Fold 1/16 and log2(e) into t once: exp(x_i*x_j/16) == exp2(t2 * x_j).
    const float t2    = sx[row0 + mi] * (0.0625f * LOG2E);

    // B-matrix selectors: column 0 <- x chunk, column 1 <- ones, else 0.
    const float sel0 = (mi == 0) ? 1.0f : 0.0f;
    const float sel1 = (mi == 1) ? 1.0f : 0.0f;

    v8f c0 = {};   // fp32 accumulators: D[:,0]=numerator, D[:,1]=denominator
    v8f c1 = {};   // second accumulator breaks the WMMA D->C serial chain

    #pragma unroll 2
    for (int j0 = 0; j0 < L_DIM; j0 += 8) {
        // Each lane fetches its K-pairs (8B aligned -> ds_load_2addr_b64,
        // 2 distinct addresses per wave -> LDS broadcast, conflict-free).
        const float2 xp0 = *(const float2*)&sx[j0     + khalf * 2];
        const float2 xp1 = *(const float2*)&sx[j0 + 4 + khalf * 2];

        v2f a0, a1, bm0, bm1;
        a0.x = __builtin_amdgcn_exp2f(t2 * xp0.x);   // v_exp_f32, TRANS pipe
        a0.y = __builtin_amdgcn_exp2f(t2 * xp0.y);
        a1.x = __builtin_amdgcn_exp2f(t2 * xp1.x);
        a1.y = __builtin_amdgcn_exp2f(t2 * xp1.y);

        bm0.x = sel0 * xp0.x + sel1;   // col0: x, col1: 1, others: 0
        bm0.y = sel0 * xp0.y + sel1;
        bm1.x = sel0 * xp1.x + sel1;
        bm1.y = sel0 * xp1.y + sel1;

        // D = A(16x4) x B(4x16) + C ; exact fp32 accumulation on matrix pipe.
        c0 = __builtin_amdgcn_wmma_f32_16x16x4_f32(
                 false, a0, false, bm0, (short)0, c0, false, false);
        c1 = __builtin_amdgcn_wmma_f32_16x16x4_f32(
                 false, a1, false, bm1, (short)0, c1, false, false);
    }

    const v8f c = c0 + c1;

    // D layout (16x16 f32): lane n (n<16) VGPR r = D[r][n]; lane n+16 VGPR r = D[8+r][n].
    // Column 0 (lanes 0,16) = numerators; column 1 (lanes 1,17) = denominators.
    if (mi < 2) {
        float* red = &sred[wave][0];
        const int base = mi * 16 + khalf * 8;   // num -> [0..15], den -> [16..31]
        red[base + 0] = c[0]; red[base + 1] = c[1];
        red[base + 2] = c[2]; red[base + 3] = c[3];
        red[base + 4] = c[4]; red[base + 5] = c[5];
        red[base + 6] = c[6]; red[base + 7] = c[7];
    }
    __syncthreads();

    if (lane < 16) {
        const float num = sred[wave][lane];
        const float den = sred[wave][16 + lane];
        y[(size_t)b * L_DIM + row0 + lane] = num / den;
    }
}

extern "C" void kernel_launch(void* const* d_in, const int* in_sizes, int n_in,
                              void* d_out, int out_size, void* d_ws, size_t ws_size,
                              hipStream_t stream) {
    const float* x = (const float*)d_in[0];
    float* y = (float*)d_out;
    const int B = in_sizes[0] / L_DIM;   // 64
    const int grid = B * (L_DIM / ROWS_PER_BLOCK);   // 1024 blocks
    rank1_softmax_attn_kernel<<<grid, 256, 0, stream>>>(x, y, B);
}